// MAUCHLoss_11089605558411
// MI455X (gfx1250) — compile-verified
//
#include <hip/hip_runtime.h>

#define NCAT  15
#define NSLOT 48   // 15 sig*lab + 15 sig + 15 lab + 1 softplus-sum, padded to 48

typedef float v2f __attribute__((ext_vector_type(2)));
typedef float f4  __attribute__((ext_vector_type(4)));
typedef float v8f __attribute__((ext_vector_type(8)));

#define LOG2E 1.4426950408889634f

// exp(-x) via the raw hardware exp2 (no libm range-fixup code)
__device__ __forceinline__ float fast_exp_neg(float x) {
  return __builtin_amdgcn_exp2f(x * -LOG2E);
}

// softplus(-t) = log(1 + exp(-t)) for t in [0,1] (t is a sigmoid output).
// Degree-6 Taylor about t=0.5; |abs err| ~1e-6 on [0,1], below the fp32
// accumulation noise of a 31M-element sum. Replaces exp+log (2 quarter-rate
// trans ops) with 6 full-rate FMAs.
__device__ __forceinline__ float softplus_neg01(float t) {
  float u = t - 0.5f;
  float p = fmaf(u, 1.884290e-4f, 8.730490e-4f);
  p = fmaf(u, p, -4.014920e-3f);
  p = fmaf(u, p, -9.592800e-3f);
  p = fmaf(u, p,  1.1750186e-1f);
  p = fmaf(u, p, -3.7754067e-1f);
  p = fmaf(u, p,  4.7407698e-1f);
  return p;
}

// Full 32-lane sum using V_WMMA_F32_16X16X4_F32 (exact fp32 path).
// A (16x4 f32): VGPR0 lanes0-15 -> A[lane][0], lanes16-31 -> A[lane-16][2];
// VGPR1 holds K=1/K=3. With a0=v, a1=0, B=ones: D[m][n] = v(m) + v(m+16).
// Sum D's 8 VGPRs per lane (rows of this half) then xor-16 merges halves.
__device__ __forceinline__ float wave_sum32(float v) {
  v2f a; a.x = v;    a.y = 0.0f;
  v2f b; b.x = 1.0f; b.y = 1.0f;
  v8f c = {};
  v8f d = __builtin_amdgcn_wmma_f32_16x16x4_f32(
      false, a, false, b, (short)0, c, false, false);
  float s = ((d[0] + d[1]) + (d[2] + d[3])) + ((d[4] + d[5]) + (d[6] + d[7]));
  s += __shfl_xor(s, 16, 32);
  return s;
}

// Pass 1: streaming partials. Each thread-iteration consumes a 4-row group
// (60 floats = 240 B, 16 B aligned) as 15 b128 non-temporal loads per array.
// Category of flat element e in a group is e % 15 (static, since 60 % 15 == 0).
__global__ __launch_bounds__(256) void mauch_partials(
    const float* __restrict__ x, const float* __restrict__ lab,
    float* __restrict__ ws, int rows) {
  float acc_sl[NCAT], acc_sig[NCAT], acc_lab[NCAT];
  float acc_sp = 0.0f;   // sum of softplus(-sig) = -sum log(s2)
#pragma unroll
  for (int c = 0; c < NCAT; ++c) { acc_sl[c] = 0.f; acc_sig[c] = 0.f; acc_lab[c] = 0.f; }

  const f4* __restrict__ x4 = (const f4*)x;
  const f4* __restrict__ l4 = (const f4*)lab;
  const int groups = rows >> 2;            // 4 rows per group
  const int stride = gridDim.x * blockDim.x;
  const int gid = blockIdx.x * blockDim.x + threadIdx.x;

  for (int g = gid; g < groups; g += stride) {
    const size_t base = (size_t)g * 15;    // in f4 units
#pragma unroll
    for (int i = 0; i < 15; ++i) {
      f4 xv = __builtin_nontemporal_load(x4 + base + i);
      f4 lv = __builtin_nontemporal_load(l4 + base + i);
#pragma unroll
      for (int j = 0; j < 4; ++j) {
        const int c = (i * 4 + j) % NCAT;
        float sig = __builtin_amdgcn_rcpf(1.0f + fast_exp_neg(xv[j]));
        acc_sig[c] += sig;
        acc_sl[c]   = fmaf(sig, lv[j], acc_sl[c]);
        acc_lab[c] += lv[j];
        acc_sp     += softplus_neg01(sig);
      }
    }
  }
  // Tail rows (rows % 4) — unused for B = 2^21 but kept general.
  const int rem_start = groups << 2;
  if (gid < rows - rem_start) {
    const size_t off = (size_t)(rem_start + gid) * NCAT;
#pragma unroll
    for (int c = 0; c < NCAT; ++c) {
      float xv = __builtin_nontemporal_load(x + off + c);
      float lv = __builtin_nontemporal_load(lab + off + c);
      float sig = __builtin_amdgcn_rcpf(1.0f + fast_exp_neg(xv));
      acc_sig[c] += sig;
      acc_sl[c]   = fmaf(sig, lv, acc_sl[c]);
      acc_lab[c] += lv;
      acc_sp     += softplus_neg01(sig);
    }
  }

  __shared__ float lds[8][NSLOT];
  const int lane = threadIdx.x & 31;
  const int wave = threadIdx.x >> 5;
#pragma unroll
  for (int c = 0; c < NCAT; ++c) {
    float s0 = wave_sum32(acc_sl[c]);   // WMMA: all lanes active here
    float s1 = wave_sum32(acc_sig[c]);
    float s2 = wave_sum32(acc_lab[c]);
    if (lane == 0) {
      lds[wave][c]          = s0;
      lds[wave][NCAT + c]   = s1;
      lds[wave][2*NCAT + c] = s2;
    }
  }
  {
    float s3 = wave_sum32(acc_sp);
    if (lane == 0) lds[wave][45] = s3;
  }
  __syncthreads();
  if (threadIdx.x < 46) {
    float t = 0.f;
#pragma unroll
    for (int w = 0; w < 8; ++w) t += lds[w][threadIdx.x];
    ws[(size_t)blockIdx.x * NSLOT + threadIdx.x] = t;
  }
}

// Pass 2: fold block partials (8-way parallel per slot), finish scalar math.
__global__ __launch_bounds__(384) void mauch_finalize(
    const float* __restrict__ ws, int nblocks, int rows, float* __restrict__ out) {
  __shared__ float part[8][NSLOT];
  __shared__ float tot[NSLOT];
  const int t = threadIdx.x;
  const int slot  = t % NSLOT;
  const int chunk = t / NSLOT;           // 0..7
  if (t < 8 * NSLOT) {
    float s = 0.f;
    for (int b = chunk; b < nblocks; b += 8) s += ws[(size_t)b * NSLOT + slot];
    part[chunk][slot] = s;
  }
  __syncthreads();
  if (t < NSLOT) {
    float s = 0.f;
#pragma unroll
    for (int p = 0; p < 8; ++p) s += part[p][t];
    tot[t] = s;
  }
  __syncthreads();
  if (t == 0) {
    const float Bf = (float)rows;
    float sum_term = 0.f, pen_last = 0.f, sum_neg_total = 0.f;
    for (int c = 0; c < NCAT; ++c) {
      float S_sl  = tot[c];              // sum sig*label
      float S_sig = tot[NCAT + c];       // sum sig
      float npos  = tot[2*NCAT + c];     // sum label
      float nneg  = Bf - npos;
      float S_neg = S_sig - S_sl;        // sum sig*(1-label)
      sum_neg_total += S_neg;
      float mean_pos = (npos > 0.f) ? S_sl  / fmaxf(npos, 1.f) : 0.f;
      float mean_neg = (nneg > 0.f) ? S_neg / fmaxf(nneg, 1.f) : 0.f;
      float pen = 1.f - mean_pos + mean_neg;
      sum_term += pen;
      if (c == NCAT - 1) pen_last = pen;
    }
    // sum(log s2) = -SP;  CEL = -(sum(Lp) - sum((1-label)sig))/N = (SP + sum_neg)/N
    const float SP = tot[45];
    const float n_elems = Bf * (float)NCAT;
    float cel = (SP + sum_neg_total) / n_elems;
    out[0] = cel + 0.1f * (sum_term / 15.0f);
    out[1] = 0.1f * pen_last;
  }
}

extern "C" void kernel_launch(void* const* d_in, const int* in_sizes, int n_in,
                              void* d_out, int out_size, void* d_ws, size_t ws_size,
                              hipStream_t stream) {
  (void)n_in; (void)out_size;
  const float* x   = (const float*)d_in[0];
  const float* lab = (const float*)d_in[1];
  const int rows = in_sizes[0] / NCAT;   // B = 2097152

  int nb = 2048;                          // 2048 blocks * 256 thr -> 1 group (4 rows)/thread
  size_t cap = ws_size / (NSLOT * sizeof(float));
  if ((size_t)nb > cap) nb = (int)cap;
  if (nb < 1) nb = 1;

  mauch_partials<<<nb, 256, 0, stream>>>(x, lab, (float*)d_ws, rows);
  mauch_finalize<<<1, 384, 0, stream>>>((const float*)d_ws, nb, rows, (float*)d_out);
}